// LossOD_K_36464272343488
// MI455X (gfx1250) — compile-verified
//
#include <hip/hip_runtime.h>
#include <math.h>

// ---------------- constants from the reference ----------------
#define FLOAT_EPS 1.1920928955078125e-07f
#define NEG_RATIO 3
#define V_XY 0.1f
#define V_WH 0.2f

#define TPB    256      // 8 wave32 waves per block
#define NCHUNK 16       // A split into 16 chunks per batch for passes 1/3
#define HBINS  2048     // LDS histogram bins for radix select
#define MAXA   16896    // LDS staging capacity (>= A=16800); 2*MAXA*4B = 132 KB

// ---------------- device helpers ----------------
__device__ __forceinline__ float smooth_l1(float x) {
    float a = fabsf(x);
    return (a < 1.0f) ? (0.5f * a * a) : (a - 0.5f);
}

// deterministic in-block tree reduction (LDS)
__device__ __forceinline__ float block_reduce(float v, float* sbuf) {
    int t = threadIdx.x;
    sbuf[t] = v;
    __syncthreads();
    for (int s = TPB / 2; s > 0; s >>= 1) {
        if (t < s) sbuf[t] += sbuf[t + s];
        __syncthreads();
    }
    float r = sbuf[0];
    __syncthreads();
    return r;
}

// loss_labels = logsumexp(p) - p[g]  (stable; C is tiny)
__device__ __forceinline__ float label_loss(const float* __restrict__ plp, int g, int C) {
    float m = plp[0];
    for (int c = 1; c < C; ++c) m = fmaxf(m, plp[c]);
    float s = 0.0f;
    for (int c = 0; c < C; ++c) s += expf(plp[c] - m);
    return m + logf(s) - plp[g];
}

// 64-bit rank key: (float bits of labels_neg desc, index asc). labels_neg >= 0,
// so its IEEE bits are monotone as unsigned. Keys are unique per anchor.
__device__ __forceinline__ unsigned long long neg_key_bits(unsigned vbits, unsigned a) {
    return ((unsigned long long)vbits << 32) | (unsigned)(~a);
}
__device__ __forceinline__ unsigned long long neg_key(float ln, unsigned a) {
    return neg_key_bits(__float_as_uint(ln), a);
}

// ---------------- pass 1: per-anchor losses -> per-(batch,chunk) partials ----------------
__global__ void pass1_stats(const float4* __restrict__ p_bboxs,   // (B,A,4)
                            const float4* __restrict__ g_bboxs,   // (B,A,4) ltrb
                            const float*  __restrict__ p_labels,  // (B,A,C)
                            const int*    __restrict__ g_labels,  // (B,A)
                            const float*  __restrict__ p_kp,      // (B,A,K)
                            const float*  __restrict__ g_kp,      // (B,A,K)
                            const float4* __restrict__ anc,       // (A,4) xywh
                            int A, int C, int K, int chunk,
                            float* __restrict__ bbox_part,
                            float* __restrict__ kp_part,
                            float* __restrict__ poslab_part,
                            float* __restrict__ cnt_part) {
    __shared__ float sred[TPB];
    const int b  = blockIdx.y;
    const int ch = blockIdx.x;
    const int a0 = ch * chunk;
    const int a1 = min(a0 + chunk, A);
    const int K2 = K >> 1;   // K is even (K=10); float2 is 8B-aligned at stride 40B

    float accB = 0.0f, accK = 0.0f, accL = 0.0f, accC = 0.0f;

    for (int a = a0 + threadIdx.x; a < a1; a += TPB) {
        const size_t ba = (size_t)b * A + a;
        // prefetch the big streams one stride ahead (gfx1250 global_prefetch_b8;
        // speculative prefetch past the end is dropped by HW)
        __builtin_prefetch(g_kp + (ba + TPB) * K, 0, 1);
        __builtin_prefetch(p_kp + (ba + TPB) * K, 0, 1);

        const int  g   = g_labels[ba];
        const bool pos = (g > 0);

        const float* plp = p_labels + ba * C;
        const float  ll  = label_loss(plp, g, C);

        if (pos) {
            accL += ll;
            accC += 1.0f;

            const float4 an  = anc[a];            // x, y, w, h
            const float4 gbb = g_bboxs[ba];       // l, t, r, b
            const float gx = (gbb.x + gbb.z) * 0.5f;
            const float gy = (gbb.y + gbb.w) * 0.5f;
            const float gw = gbb.z - gbb.x;
            const float gh = gbb.w - gbb.y;

            const float sx = V_XY * an.z;         // V_XY * anc_w
            const float sy = V_XY * an.w;         // V_XY * anc_h
            const float d0 = (gx - an.x) / sx;
            const float d1 = (gy - an.y) / sy;
            const float d2 = logf(gw / an.z) / V_WH;
            const float d3 = logf(gh / an.w) / V_WH;

            const float4 pbb = p_bboxs[ba];
            accB += smooth_l1(pbb.x - d0) + smooth_l1(pbb.y - d1) +
                    smooth_l1(pbb.z - d2) + smooth_l1(pbb.w - d3);

            // keypoints: anchor (x,y)/(w,h) tiled across the K/2 points
            const float2* gk2 = (const float2*)(g_kp + ba * K);
            const float2* pk2 = (const float2*)(p_kp + ba * K);
            float skp = 0.0f;
            bool  allp = true;
            for (int j = 0; j < K2; ++j) {
                const float2 gp = gk2[j];
                const float2 pp = pk2[j];
                allp = allp && (gp.x > 0.0f) && (gp.y > 0.0f);
                const float dx = (gp.x - an.x) / sx;
                const float dy = (gp.y - an.y) / sy;
                skp += smooth_l1(pp.x - dx) + smooth_l1(pp.y - dy);
            }
            if (allp) accK += skp;
        }
    }

    const float rB = block_reduce(accB, sred);
    const float rK = block_reduce(accK, sred);
    const float rL = block_reduce(accL, sred);
    const float rC = block_reduce(accC, sred);
    if (threadIdx.x == 0) {
        const int idx = b * gridDim.x + ch;
        bbox_part[idx]   = rB;
        kp_part[idx]     = rK;
        poslab_part[idx] = rL;
        cnt_part[idx]    = rC;
    }
}

// Parallel choice of the histogram bin holding the k-th largest key.
// Reads sh_k/prefix BEFORE the scan barriers; the unique winning thread writes
// sh_k/sh_prefix AFTER them, so no read/write race. hist[] must be synced.
__device__ __forceinline__ void choose_bin_parallel(unsigned* __restrict__ hist, int nb,
                                                    int shift, unsigned long long prefix,
                                                    unsigned* __restrict__ sscan,
                                                    int* __restrict__ sh_k,
                                                    unsigned long long* __restrict__ sh_prefix) {
    const int t   = threadIdx.x;
    const int per = nb / TPB;          // 8 (2048 bins) or 2 (512 bins)
    const int base = t * per;
    const int k = *sh_k;               // read before any writer can run (barriers below)

    unsigned cs = 0;
    for (int i = 0; i < per; ++i) cs += hist[base + i];
    sscan[t] = cs;
    __syncthreads();
    // Hillis-Steele inclusive suffix scan over the 256 chunk sums
    for (int off = 1; off < TPB; off <<= 1) {
        const unsigned v = (t + off < TPB) ? sscan[t + off] : 0u;
        __syncthreads();
        sscan[t] += v;
        __syncthreads();
    }
    const unsigned above_chunk = sscan[t] - cs;   // keys in bins above my chunk

    unsigned cum = above_chunk;                   // keys strictly above current bin
    for (int i = per - 1; i >= 0; --i) {
        const unsigned hc = hist[base + i];
        if (cum < (unsigned)k && cum + hc >= (unsigned)k) {   // exactly one (t,i)
            *sh_k = k - (int)cum;
            *sh_prefix = prefix | ((unsigned long long)(base + i) << shift);
        }
        cum += hc;
    }
    __syncthreads();
}

// ---------------- pass 2 (CDNA5 big-LDS path): stage labels_neg/loss in LDS once,
// radix-select the exact pivot, and sum the selected losses — all from LDS. ----------------
__global__ void select_and_neg_lds(const float* __restrict__ p_labels,
                                   const int*   __restrict__ g_labels,
                                   const float* __restrict__ cnt_part,
                                   int A, int C, int nchunks,
                                   int*   __restrict__ pos_num_out,
                                   float* __restrict__ neg_sum_out) {
    __shared__ unsigned           lnbits[MAXA];   // float bits of labels_neg (>= 0)
    __shared__ float              lossv[MAXA];    // loss_labels
    __shared__ unsigned           hist[HBINS];
    __shared__ unsigned           sscan[TPB];
    __shared__ float              sred[TPB];
    __shared__ unsigned long long sh_prefix;
    __shared__ int                sh_k;

    const int b = blockIdx.x;
    const int t = threadIdx.x;

    // compute per-anchor CE loss ONCE; stage in LDS (global reads happen exactly once)
    for (int a = t; a < A; a += TPB) {
        const size_t ba = (size_t)b * A + a;
        const int    g  = g_labels[ba];
        const float  ll = label_loss(p_labels + ba * C, g, C);
        const float  ln = (g > 0) ? 0.0f : ll;
        lnbits[a] = __float_as_uint(ln);
        lossv[a]  = ll;
    }

    // reduce positive count for this batch from pass-1 partials
    float c = 0.0f;
    for (int i = t; i < nchunks; i += TPB) c += cnt_part[b * nchunks + i];
    c = block_reduce(c, sred);                 // also a full barrier: LDS staging visible
    const int pos_num = (int)(c + 0.5f);
    const int neg_num = min(NEG_RATIO * pos_num, A);
    if (t == 0) pos_num_out[b] = pos_num;

    if (neg_num <= 0) {                        // no negatives selected
        if (t == 0) neg_sum_out[b] = 0.0f;
        return;
    }

    unsigned long long pv = 0ull;              // neg_num >= A -> select everything
    if (neg_num < A) {
        if (t == 0) { sh_prefix = 0ull; sh_k = neg_num; }
        __syncthreads();

        const int shifts[6] = {53, 42, 31, 20, 9, 0};
        const int widths[6] = {11, 11, 11, 11, 11, 9};

        for (int lv = 0; lv < 6; ++lv) {
            const int shift = shifts[lv];
            const int nb    = 1 << widths[lv];
            for (int i = t; i < nb; i += TPB) hist[i] = 0u;
            __syncthreads();

            const unsigned long long prefix = sh_prefix;
            const int bound = shift + widths[lv];
            const unsigned long long bmask = (bound >= 64) ? 0ull : (~0ull << bound);

            for (int a = t; a < A; a += TPB) {
                const unsigned long long key = neg_key_bits(lnbits[a], (unsigned)a);
                if ((key & bmask) == prefix)
                    atomicAdd(&hist[(unsigned)((key >> shift) & (unsigned)(nb - 1))], 1u);
            }
            __syncthreads();

            choose_bin_parallel(hist, nb, shift, prefix, sscan, &sh_k, &sh_prefix);
        }
        pv = sh_prefix;                        // exact 64-bit key of neg_num-th largest
    }

    // selected-negative sum, straight out of LDS
    float acc = 0.0f;
    for (int a = t; a < A; a += TPB) {
        const unsigned long long key = neg_key_bits(lnbits[a], (unsigned)a);
        if (key >= pv) acc += lossv[a];        // mask_neg * loss_labels (positives too)
    }
    acc = block_reduce(acc, sred);
    if (t == 0) neg_sum_out[b] = acc;
}

// ---------------- fallback pass 2: radix select recomputing keys from global ----------------
__global__ void select_pivot(const float* __restrict__ p_labels,
                             const int*   __restrict__ g_labels,
                             const float* __restrict__ cnt_part,
                             int A, int C, int nchunks,
                             unsigned long long* __restrict__ pivot,
                             int* __restrict__ pos_num_out) {
    __shared__ unsigned           hist[HBINS];
    __shared__ unsigned           sscan[TPB];
    __shared__ float              sred[TPB];
    __shared__ unsigned long long sh_prefix;
    __shared__ int                sh_k;

    const int b = blockIdx.x;
    const int t = threadIdx.x;

    float c = 0.0f;
    for (int i = t; i < nchunks; i += TPB) c += cnt_part[b * nchunks + i];
    c = block_reduce(c, sred);
    const int pos_num = (int)(c + 0.5f);
    const int neg_num = min(NEG_RATIO * pos_num, A);
    if (t == 0) pos_num_out[b] = pos_num;

    if (neg_num <= 0) { if (t == 0) pivot[b] = ~0ull; return; }
    if (neg_num >= A) { if (t == 0) pivot[b] = 0ull;  return; }

    if (t == 0) { sh_prefix = 0ull; sh_k = neg_num; }
    __syncthreads();

    const int shifts[6] = {53, 42, 31, 20, 9, 0};
    const int widths[6] = {11, 11, 11, 11, 11, 9};

    for (int lv = 0; lv < 6; ++lv) {
        const int shift = shifts[lv];
        const int nb    = 1 << widths[lv];
        for (int i = t; i < nb; i += TPB) hist[i] = 0u;
        __syncthreads();

        const unsigned long long prefix = sh_prefix;
        const int bound = shift + widths[lv];
        const unsigned long long bmask = (bound >= 64) ? 0ull : (~0ull << bound);

        for (int a = t; a < A; a += TPB) {
            const size_t ba = (size_t)b * A + a;
            const int    g  = g_labels[ba];
            float ln = 0.0f;
            if (g <= 0) ln = label_loss(p_labels + ba * C, 0, C);
            const unsigned long long key = neg_key(ln, (unsigned)a);
            if ((key & bmask) == prefix)
                atomicAdd(&hist[(unsigned)((key >> shift) & (unsigned)(nb - 1))], 1u);
        }
        __syncthreads();

        choose_bin_parallel(hist, nb, shift, prefix, sscan, &sh_k, &sh_prefix);
    }
    if (t == 0) pivot[b] = sh_prefix;
}

// ---------------- fallback pass 3: sum loss over selected negatives ----------------
__global__ void pass3_neg(const float* __restrict__ p_labels,
                          const int*   __restrict__ g_labels,
                          const unsigned long long* __restrict__ pivot,
                          int A, int C, int chunk,
                          float* __restrict__ neg_part) {
    __shared__ float sred[TPB];
    const int b  = blockIdx.y;
    const int ch = blockIdx.x;
    const unsigned long long pv = pivot[b];
    const int a0 = ch * chunk;
    const int a1 = min(a0 + chunk, A);

    float acc = 0.0f;
    for (int a = a0 + threadIdx.x; a < a1; a += TPB) {
        const size_t ba = (size_t)b * A + a;
        const int    g  = g_labels[ba];
        const float  ll = label_loss(p_labels + ba * C, g, C);
        const float  ln = (g > 0) ? 0.0f : ll;
        const unsigned long long key = neg_key(ln, (unsigned)a);
        if (key >= pv) acc += ll;
    }
    acc = block_reduce(acc, sred);
    if (threadIdx.x == 0) neg_part[b * gridDim.x + ch] = acc;
}

// ---------------- pass 4: per-batch normalization, mean over B, 4 outputs ----------------
__global__ void finalize(const float* __restrict__ bbox_part,
                         const float* __restrict__ kp_part,
                         const float* __restrict__ poslab_part,
                         const float* __restrict__ neg_part,
                         const int*   __restrict__ pos_num,
                         int B, int nchunks, int neg_chunks,
                         float* __restrict__ out) {
    __shared__ float sb[TPB], sl[TPB], sk[TPB];
    const int t = threadIdx.x;
    float lb = 0.0f, ll = 0.0f, lk = 0.0f;
    if (t < B) {
        float vb = 0.0f, vk = 0.0f, vl = 0.0f, vn = 0.0f;
        for (int i = 0; i < nchunks; ++i) {
            vb += bbox_part[t * nchunks + i];
            vk += kp_part[t * nchunks + i];
            vl += poslab_part[t * nchunks + i];
        }
        for (int i = 0; i < neg_chunks; ++i) vn += neg_part[t * neg_chunks + i];
        const int   pn  = pos_num[t];
        const float nm  = (pn > 0) ? 1.0f : 0.0f;
        const float den = fmaxf((float)pn, FLOAT_EPS);
        lb = vb * nm / den;
        ll = (vl + vn) * nm / den;
        lk = vk * nm / den;
    }
    sb[t] = lb; sl[t] = ll; sk[t] = lk;
    __syncthreads();
    for (int s = TPB / 2; s > 0; s >>= 1) {
        if (t < s) { sb[t] += sb[t + s]; sl[t] += sl[t + s]; sk[t] += sk[t + s]; }
        __syncthreads();
    }
    if (t == 0) {
        const float invB = 1.0f / (float)B;
        const float mb = sb[0] * invB, mlab = sl[0] * invB, mk = sk[0] * invB;
        out[0] = mb + mlab + mk;   // LOSS_WEIGHT = (1,1,1)
        out[1] = mb;
        out[2] = mlab;
        out[3] = mk;
    }
}

// ---------------- host launch ----------------
extern "C" void kernel_launch(void* const* d_in, const int* in_sizes, int n_in,
                              void* d_out, int out_size, void* d_ws, size_t ws_size,
                              hipStream_t stream) {
    const float* p_bboxs  = (const float*)d_in[0];
    const float* g_bboxs  = (const float*)d_in[1];
    const float* p_labels = (const float*)d_in[2];
    const int*   g_labels = (const int*)d_in[3];
    const float* p_kp     = (const float*)d_in[4];
    const float* g_kp     = (const float*)d_in[5];
    const float* anc      = (const float*)d_in[6];

    const int A = in_sizes[6] / 4;                 // anc is (A,4)
    const int B = in_sizes[3] / A;                 // g_labels is (B,A)
    const int C = in_sizes[2] / (B * A);           // p_labels is (B,A,C)
    const int K = in_sizes[4] / (B * A);           // p_keypoints is (B,A,K)
    const int chunk = (A + NCHUNK - 1) / NCHUNK;

    // workspace layout (8-byte aligned first): ~22 KB total
    char* ws = (char*)d_ws;
    unsigned long long* pivot = (unsigned long long*)ws; ws += (size_t)B * sizeof(unsigned long long);
    int*   pos_num     = (int*)ws;   ws += (size_t)B * sizeof(int);
    float* bbox_part   = (float*)ws; ws += (size_t)B * NCHUNK * sizeof(float);
    float* kp_part     = (float*)ws; ws += (size_t)B * NCHUNK * sizeof(float);
    float* poslab_part = (float*)ws; ws += (size_t)B * NCHUNK * sizeof(float);
    float* cnt_part    = (float*)ws; ws += (size_t)B * NCHUNK * sizeof(float);
    float* neg_part    = (float*)ws;               // [B][NCHUNK] or [B]

    dim3 grid(NCHUNK, B);
    pass1_stats<<<grid, TPB, 0, stream>>>((const float4*)p_bboxs, (const float4*)g_bboxs,
                                          p_labels, g_labels, p_kp, g_kp,
                                          (const float4*)anc, A, C, K, chunk,
                                          bbox_part, kp_part, poslab_part, cnt_part);
    if (A <= MAXA) {
        // CDNA5 path: 320 KB WGP LDS lets one workgroup stage all per-anchor CE
        // losses + rank keys; one kernel does select + selected-sum from LDS.
        select_and_neg_lds<<<B, TPB, 0, stream>>>(p_labels, g_labels, cnt_part,
                                                  A, C, NCHUNK, pos_num, neg_part);
        finalize<<<1, TPB, 0, stream>>>(bbox_part, kp_part, poslab_part, neg_part,
                                        pos_num, B, NCHUNK, 1, (float*)d_out);
    } else {
        select_pivot<<<B, TPB, 0, stream>>>(p_labels, g_labels, cnt_part, A, C, NCHUNK,
                                            pivot, pos_num);
        pass3_neg<<<grid, TPB, 0, stream>>>(p_labels, g_labels, pivot, A, C, chunk, neg_part);
        finalize<<<1, TPB, 0, stream>>>(bbox_part, kp_part, poslab_part, neg_part,
                                        pos_num, B, NCHUNK, NCHUNK, (float*)d_out);
    }
}